// CallaghanRestrictedCylinder_1554778161274
// MI455X (gfx1250) — compile-verified
//
#include <hip/hip_runtime.h>
#include <math.h>
#include <stdint.h>

// Callaghan restricted-cylinder signal on MI455X (gfx1250, wave32).
//
// Roofline: ~2 MB total HBM traffic (~0.1 us @ 23.3 TB/s) vs ~1 GFLOP f32 +
// 1e8 v_rcp_f32 -> VALU/transcendental bound. Strategy:
//  * precompute the 1000 (root,order) weights once (kernel 1),
//  * one thread per sample; the 8 KB of per-(r,m) constants are staged into
//    LDS once per workgroup by the Tensor Data Mover (tensor_load_to_lds +
//    s_wait_tensorcnt) -- CDNA5 async DMA path instead of per-thread loads,
//  * Miller backward Bessel recurrence fused with the term accumulation:
//    all terms are quadratic in J, so accumulate unnormalized and apply
//    inv(S)^2 at the end (conditional rescale keeps f64 in range),
//  * v_rcp_f32 for the 1000 per-sample denominators (no IEEE div sequences),
//  * the bvecs @ mu matvec runs on V_WMMA_F32_16X16X4_F32 (f32 tensor path).

#define N_ROOTS 20
#define N_FUNCS 50
#define NRM (N_ROOTS * N_FUNCS)

typedef __attribute__((ext_vector_type(2))) float        v2f;
typedef __attribute__((ext_vector_type(8))) float        v8f;
typedef __attribute__((ext_vector_type(4))) unsigned int v4u;
typedef __attribute__((ext_vector_type(4))) int          v4i;
typedef __attribute__((ext_vector_type(8))) int          v8i;

#if __has_builtin(__builtin_amdgcn_tensor_load_to_lds)
#define HAVE_TDM 1
#if __has_include(<hip/amd_detail/amd_gfx1250_TDM.h>)
#define TDM_SIX_ARGS 1   // therock-10.0 headers -> clang-23 6-arg builtin
#endif
#endif

// ---------------------------------------------------------------------------
// Kernel 1: per-(root, order) constants -> workspace (m-major: idx = m*20 + r)
//   ws[0..999]    : a2[r,m]   = alpha^2
//   ws[1000..1999]: w[r,m]    = 8*exp(-a2*Dp*tau/R^2)                   (m==0)
//                             = 16*exp(-a2*Dp*tau/R^2) * a2/(a2 - m^2)  (m>=1)
// ---------------------------------------------------------------------------
__global__ __launch_bounds__(256) void callaghan_precomp(
    const float* __restrict__ alpha,   // (N_ROOTS, N_FUNCS) row-major
    const float* __restrict__ p_diam,
    const float* __restrict__ p_dperp,
    const float* __restrict__ p_tau,
    float* __restrict__ ws)            // 2*NRM floats, contiguous
{
    int t = blockIdx.x * blockDim.x + threadIdx.x;
    if (t >= NRM) return;
    int r = t / N_FUNCS;
    int m = t - r * N_FUNCS;
    float a  = alpha[t];
    float a2 = a * a;
    float radius = 0.5f * p_diam[0];
    float et = expf(-a2 * p_dperp[0] * p_tau[0] / (radius * radius));
    float w  = (m == 0) ? (8.0f * et)
                        : (16.0f * et * a2 / (a2 - (float)(m * m)));
    int idx = m * N_ROOTS + r;   // m-major so the inner r-loop is contiguous
    ws[idx]       = a2;
    ws[NRM + idx] = w;
}

// ---------------------------------------------------------------------------
// Kernel 2: main fused kernel, one thread per sample.
// ---------------------------------------------------------------------------
__global__ __launch_bounds__(256) void callaghan_main(
    const float* __restrict__ bvals,
    const float* __restrict__ bvecs,   // (N,3)
    const float* __restrict__ mu,      // [theta, phi]
    const float* __restrict__ p_lam,
    const float* __restrict__ p_diam,
    const float* __restrict__ p_tau,
    const float* __restrict__ ws,      // 2*NRM floats (a2 | w), contiguous
    float* __restrict__ out,
    int n)
{
    __shared__ float sConst[2 * NRM];  // 8 KB: [0..999]=a2, [1000..1999]=w
    __shared__ float sDot[256];        // WMMA D-tile spill for per-lane dots
    const float* sA2 = sConst;
    const float* sW  = sConst + NRM;

    const int tid = threadIdx.x;

    // ---- Stage constants into LDS via the Tensor Data Mover ----------------
    // 1D tensor: 2000 x f32 contiguous, one DMA issued by wave 0, tracked on
    // TENSORcnt. D# per CDNA5 ISA ch.8 (group0: count/lds/global/type=2;
    // group1: data_size=4B, tensor_dim0=tile_dim0=2000, stride=2000).
#ifdef HAVE_TDM
    if (tid == 0) {
        unsigned           lds_off = (unsigned)(uintptr_t)sConst; // flat->LDS low 32b
        unsigned long long ga      = (unsigned long long)(uintptr_t)ws;
        v4u g0;
        g0.x = 1u;                                        // count=1, user mode
        g0.y = lds_off;                                   // lds_addr
        g0.z = (unsigned)ga;                              // global_addr[31:0]
        g0.w = (unsigned)((ga >> 32) & 0x01FFFFFFu)       // global_addr[56:32]
             | (2u << 30);                                // type = 2 ("image")
        v8i g1;
        g1[0] = 0x00020000;                  // wg_mask=0, data_size=2 (4 bytes)
        g1[1] = (int)((unsigned)(2 * NRM) << 16);  // tensor_dim0[15:0] @bits63:48
        g1[2] = (int)(1u << 16);             // tensor_dim0 hi=0 | tensor_dim1=1
        g1[3] = (int)((unsigned)(2 * NRM) << 16);  // dim1 hi=0 | tile_dim0=2000
        g1[4] = 1;                           // tile_dim1=1, tile_dim2=0
        g1[5] = 2 * NRM;                     // tensor_dim0_stride lo
        g1[6] = 0;                           // stride hi | dim1_stride lo
        g1[7] = 0;
        v4i gz = {0, 0, 0, 0};               // groups 2/3 unused (<=2D tensor)
#ifdef TDM_SIX_ARGS
        v8i gz8 = {0, 0, 0, 0, 0, 0, 0, 0};
        __builtin_amdgcn_tensor_load_to_lds(g0, g1, gz, gz, gz8, 0);
#else
        __builtin_amdgcn_tensor_load_to_lds(g0, g1, gz, gz, 0);
#endif
        __builtin_amdgcn_s_wait_tensorcnt(0);
    }
#else
    for (int k = tid; k < 2 * NRM; k += 256) sConst[k] = ws[k];
#endif

    // mu_cart (uniform across all threads).
    float th = mu[0], ph = mu[1];
    float st = sinf(th), ct = cosf(th);
    float sp = sinf(ph), cp = cosf(ph);
    float mux = st * cp, muy = st * sp, muz = ct;

    const int  lane  = tid & 31;
    const int  wv    = tid >> 5;
    const int  wbase = blockIdx.x * 256 + wv * 32;   // first sample of wave
    const bool hi    = lane >= 16;

    // ---- dot[n] = bvecs[n] . mu via V_WMMA_F32_16X16X4_F32 -----------------
    // A (16x4 f32): lanes 0-15 hold K=0 (v0) / K=1 (v1); lanes 16-31 K=2/K=3.
    //   row M = lane%16 = local sample; K = {x, y, z, 0}.
    // B (4x16): mu component per K, replicated across all 16 columns, so
    //   every column of D equals the dot product for that row.
#if __has_builtin(__builtin_amdgcn_wmma_f32_16x16x4_f32)
    for (int t2 = 0; t2 < 2; ++t2) {              // two 16-row tiles = 32 samples
        int srow = wbase + t2 * 16 + (lane & 15);
        float gx = 0.0f, gy = 0.0f, gz = 0.0f;
        if (srow < n) {
            gx = bvecs[3 * srow + 0];
            gy = bvecs[3 * srow + 1];
            gz = bvecs[3 * srow + 2];
        }
        v2f a;  a.x = hi ? gz  : gx;  a.y = hi ? 0.0f : gy;
        v2f b;  b.x = hi ? muz : mux; b.y = hi ? 0.0f : muy;
        v8f c = {};
        // EXEC is all-ones here (divergent load guard reconverged above).
        c = __builtin_amdgcn_wmma_f32_16x16x4_f32(
                false, a, false, b, (short)0, c, false, false);
        // D layout: VGPR j -> row M = j + (hi?8:0); all columns identical.
        if ((lane & 15) == 0) {
            int r0 = wv * 32 + t2 * 16 + (hi ? 8 : 0);
            #pragma unroll
            for (int j = 0; j < 8; ++j) sDot[r0 + j] = c[j];
        }
    }
#else
    {
        int s = wbase + lane;
        float d = 0.0f;
        if (s < n)
            d = bvecs[3*s]*mux + bvecs[3*s+1]*muy + bvecs[3*s+2]*muz;
        sDot[tid] = d;
    }
#endif

    __syncthreads();   // releases TDM-staged constants + dot tile to all waves

    const int i = blockIdx.x * 256 + tid;
    if (i >= n) return;

    const float dot = sDot[tid];
    const float bv  = bvals[i];
    const float lam = p_lam[0];
    const float tau = p_tau[0];
    const float radius = 0.5f * p_diam[0];

    const float sigpar = __expf(-bv * lam * dot * dot);

    float sin2 = 1.0f - dot * dot;
    sin2 = fminf(fmaxf(sin2, 0.0f), 1.0f);
    const float PI2 = 6.28318530717958647692f;
    // reciprocal instead of the IEEE v_div_scale/v_div_fmas sequence
    const float inv_tau = __builtin_amdgcn_rcpf(tau + 1e-12f);
    const float q_mag  = sqrtf(bv * inv_tau) * (1000.0f / PI2);
    const float q_perp = q_mag * sqrtf(sin2);
    const float x  = PI2 * q_perp * radius;     // q_arg, in [0, ~1.6]
    const float q2 = x * x;

    float res = 1.0f;
    if (q_perp > 1e-9f) {
        // m = 0 root sum (used at the m==1 step of the recurrence below).
        float s0 = 0.0f;
        #pragma unroll
        for (int r = 0; r < N_ROOTS; ++r) {
            float d  = q2 - sA2[r];
            float d2 = fmaxf(d * d, 1e-12f);
            s0 = fmaf(sW[r], __builtin_amdgcn_rcpf(d2), s0);
        }

        // Miller backward recurrence J_{m-1} = (2m/x) J_m - J_{m+1}, fused
        // with the term accumulation. All result terms are quadratic in J,
        // so accumulate with unnormalized J and scale by inv(S)^2 at the end.
        // Conditional rescale keeps the f64 magnitudes (and their squares)
        // in range for arbitrarily small x.
        const int MSTART = 64;                   // even, > 50 + margin
        const double xd = (double)x;
        const double tw = 2.0 / xd;
        double jp  = 0.0;                        // J_{m+1} (raw)
        double jc  = 1e-30;                      // J_m     (raw)
        double nrm = 0.0;                        // S = J0 + 2*sum_{even} J_m
        double acc = 0.0;                        // raw (scale^2) result
        for (int m = MSTART; m >= 1; --m) {
            double jm = (double)m * tw * jc - jp;   // J_{m-1} (raw)
            if ((m & 1) == 0) nrm += jc + jc;       // even-order normalizer
            if (m <= N_FUNCS - 1) {
                // sum over roots for this order (LDS broadcast reads)
                float sm = 0.0f;
                const int base = m * N_ROOTS;
                #pragma unroll
                for (int r = 0; r < N_ROOTS; ++r) {
                    float d  = q2 - sA2[base + r];
                    float d2 = fmaxf(d * d, 1e-12f);
                    sm = fmaf(sW[base + r], __builtin_amdgcn_rcpf(d2), sm);
                }
                double g = 0.5 * xd * (jm - jp);    // x * J'_m (raw scale)
                acc += (double)sm * (g * g);
            }
            if (m == 1) {
                // m = 0 term: 8*exp*q^2/denom summed, * J1^2 ; jc == J_1 raw
                acc += (double)s0 * (double)q2 * (jc * jc);
            }
            if (fabs(jm) > 1e100) {                 // rare rescale
                const double sc = 1e-100;
                jm  *= sc; jc  *= sc;
                nrm *= sc; acc *= sc * sc;
            }
            jp = jc; jc = jm;
        }
        nrm += jc;                                  // + J_0
        double inv = 1.0 / nrm;
        res = (float)(acc * inv * inv);
    }

    out[i] = sigpar * res;
}

// ---------------------------------------------------------------------------
extern "C" void kernel_launch(void* const* d_in, const int* in_sizes, int n_in,
                              void* d_out, int out_size, void* d_ws, size_t ws_size,
                              hipStream_t stream) {
    const float* bvals = (const float*)d_in[0];
    const float* bvecs = (const float*)d_in[1];
    const float* mu    = (const float*)d_in[2];
    const float* lam   = (const float*)d_in[3];
    const float* diam  = (const float*)d_in[4];
    const float* dperp = (const float*)d_in[5];
    const float* tau   = (const float*)d_in[6];
    const float* alpha = (const float*)d_in[7];
    float* out = (float*)d_out;
    const int n = in_sizes[0];

    // workspace: 2 * 1000 floats = 8 KB, contiguous (a2 block then w block)
    float* ws = (float*)d_ws;

    callaghan_precomp<<<(NRM + 255) / 256, 256, 0, stream>>>(
        alpha, diam, dperp, tau, ws);

    callaghan_main<<<(n + 255) / 256, 256, 0, stream>>>(
        bvals, bvecs, mu, lam, diam, tau, ws, out, n);
}